// GeneralAttention_87969520156964
// MI455X (gfx1250) — compile-verified
//
#include <hip/hip_runtime.h>

typedef __attribute__((ext_vector_type(2))) float v2f;
typedef __attribute__((ext_vector_type(8))) float v8f;

#define NC      16384   // n_chains = B*Lq*RUNS
#define STEPS   64
#define L_KV    4096
#define D_MODEL 64
#define RUNS    4

__global__ __launch_bounds__(128)
void gibbs_subset_attn_kernel(const float* __restrict__ q,
                              const float* __restrict__ k,
                              const float* __restrict__ v,
                              const int*   __restrict__ vidx,
                              const float* __restrict__ z,
                              float* __restrict__ out)
{
    __shared__ float lds_dots[4 * STEPS];    // per-wave logits
    __shared__ float lds_out[4 * D_MODEL];   // per-wave chain outputs

    const int tid  = threadIdx.x;
    const int wave = tid >> 5;
    const int lane = tid & 31;
    const int row  = blockIdx.x;            // b*Lq + lq  (grid = 4096)
    const int c    = row * RUNS + wave;     // chain id
    const int b    = row >> 10;             // Lq = 1024
    const int kh   = lane >> 4;             // lane half (K select)
    const int m    = lane & 15;             // M index within tile

    const float* qrow  = q + (size_t)row * D_MODEL;
    const float* kbase = k + (size_t)b * L_KV * D_MODEL;
    const float* vbase = v + (size_t)b * L_KV * D_MODEL;

    // ---- Phase 1: all 64 logits per chain via V_WMMA_F32_16X16X4_F32 ----
    // A tile: 16 gathered k rows (M=steps, K=64 over 16 chunks of 4).
    // B tile: q broadcast to all 16 columns (only K mapping matters).
    for (int t4 = 0; t4 < 4; ++t4) {
        const int s  = 16 * t4 + m;
        const int vi = vidx[s * NC + c];
        const float* krow = kbase + (size_t)vi * D_MODEL;
        v8f acc = {0.f, 0.f, 0.f, 0.f, 0.f, 0.f, 0.f, 0.f};
#pragma unroll
        for (int kk = 0; kk < 16; ++kk) {
            const int j = 4 * kk + 2 * kh;
            v2f a2; a2.x = krow[j]; a2.y = krow[j + 1];
            v2f b2; b2.x = qrow[j]; b2.y = qrow[j + 1];
            acc = __builtin_amdgcn_wmma_f32_16x16x4_f32(
                false, a2, false, b2, (short)0, acc, false, false);
        }
        // D[M=r+8*kh][N=lane&15]; all N identical -> lanes with N==0 write.
        if (m == 0) {
#pragma unroll
            for (int r = 0; r < 8; ++r)
                lds_dots[wave * STEPS + 16 * t4 + 8 * kh + r] = acc[r];
        }
    }

    // ---- Phase 2: sequential Gibbs scan; 4096-bit mask in 4 VGPRs/lane ----
    unsigned m0 = 0, m1 = 0, m2 = 0, m3 = 0;
    float svx = 0.f, svy = 0.f, cnt = 0.f;
    const float scale = 0.125f;             // 1/sqrt(64)

    for (int s = 0; s < STEPS; ++s) {
        const int   vi = vidx[s * NC + c];
        const float zs = z[s * NC + c];
        const float a  = lds_dots[wave * STEPS + s] * scale;
        const float p  = 1.0f / (1.0f + expf(-a));
        const bool newh = (zs <= p);        // wave-uniform

        const int ow = vi >> 7;             // owner lane (128 bits per lane)
        const int w  = (vi >> 5) & 3;       // word within owner
        const unsigned bit  = 1u << (vi & 31);
        const unsigned word = (w == 0) ? m0 : (w == 1) ? m1 : (w == 2) ? m2 : m3;
        const bool mine = (lane == ow);
        const bool oldb = (__ballot(mine && ((word & bit) != 0u)) != 0ull);
        const int sign = (int)newh - (int)oldb;   // wave-uniform in {-1,0,1}

        if (mine) {
            const unsigned nm = newh ? (word | bit) : (word & ~bit);
            if (w == 0) m0 = nm; else if (w == 1) m1 = nm;
            else if (w == 2) m2 = nm; else m3 = nm;
        }
        if (sign != 0) {                    // uniform branch; coalesced gather
            const float2 vv = *(const float2*)(vbase + (size_t)vi * D_MODEL + 2 * lane);
            const float fs = (float)sign;
            svx = fmaf(fs, vv.x, svx);
            svy = fmaf(fs, vv.y, svy);
            cnt += fs;
        }
    }

    // ---- Phase 3: mean over the 4 runs, single deterministic store ----
    const float denom = fmaxf(cnt, 1.0f);
    lds_out[wave * D_MODEL + 2 * lane]     = svx / denom;
    lds_out[wave * D_MODEL + 2 * lane + 1] = svy / denom;
    __syncthreads();

    if (tid < D_MODEL) {
        const float o = 0.25f * (lds_out[tid] +
                                 lds_out[D_MODEL + tid] +
                                 lds_out[2 * D_MODEL + tid] +
                                 lds_out[3 * D_MODEL + tid]);
        out[(size_t)row * D_MODEL + tid] = o;
    }
}

extern "C" void kernel_launch(void* const* d_in, const int* in_sizes, int n_in,
                              void* d_out, int out_size, void* d_ws, size_t ws_size,
                              hipStream_t stream) {
    const float* q    = (const float*)d_in[0];
    const float* k    = (const float*)d_in[1];
    const float* v    = (const float*)d_in[2];
    const int*   vidx = (const int*)d_in[3];
    const float* z    = (const float*)d_in[4];
    float* out = (float*)d_out;

    dim3 grid(4096);   // B*Lq workgroups, one per output row
    dim3 block(128);   // 4 wave32, one per run-chain
    hipLaunchKernelGGL(gibbs_subset_attn_kernel, grid, block, 0, stream,
                       q, k, v, vidx, z, out);
}